// PointsToMasks_21955872817177
// MI455X (gfx1250) — compile-verified
//
#include <hip/hip_runtime.h>
#include <stdint.h>

#define HH 512
#define WW 512
#define CHN 96            // B*V = 32*3
#define NPTS 128
#define NPT_TOT (CHN*NPTS)         // 12288
#define HW (HH*WW)                 // 262144
#define MASK_ELEMS ((size_t)CHN*HW) // 25,165,824
#define OOBV 1e-5f
#define LRATE 0.2f
#define ALPHAV 0.99f
#define EPSV 1e-8f

// ---- ws float layout (small state only; pyramid lives in masks_opt region) ----
#define WS_P    0          // 24576 floats (x,y per point)
#define WS_SQ   24576      // 24576 floats
#define WS_BMAX 49152      // 48 floats
#define WS_DONE 49200      // 1 int

// pyramid levels 1..9, offsets in floats inside the pyr buffer
constexpr int LSZ[10]  = {512, 256, 128, 64, 32, 16, 8, 4, 2, 1};
constexpr int LOFF[10] = {0,
                          0,          // level1: 96*256*256
                          6291456,    // level2
                          7913728, 8306944, 8405248, 8429824,
                          8435968, 8437504, 8437888};

typedef unsigned int v4u __attribute__((ext_vector_type(4)));
typedef int          v8i __attribute__((ext_vector_type(8)));
typedef int          v4i __attribute__((ext_vector_type(4)));

__device__ __forceinline__ int rnd_clamp(float v) {
  int i = __float2int_rn(v);          // round-half-even, matches jnp.round
  return min(max(i, 0), HH - 1);
}

// bilinear sample of level (size s) at output-space integer coords (iy,ix),
// align_corners=False semantics with edge clamp
__device__ __forceinline__ float bilerp(const float* p, int s, int iy, int ix) {
  float sc = (float)s * (1.0f / 512.0f);
  float fy = ((float)iy + 0.5f) * sc - 0.5f;
  float fx = ((float)ix + 0.5f) * sc - 0.5f;
  float fy0 = floorf(fy), fx0 = floorf(fx);
  float ty = fy - fy0, tx = fx - fx0;
  int y0 = (int)fy0, x0 = (int)fx0;
  int ya = min(max(y0, 0), s - 1), yb = min(max(y0 + 1, 0), s - 1);
  int xa = min(max(x0, 0), s - 1), xb = min(max(x0 + 1, 0), s - 1);
  float v00 = p[ya * s + xa], v01 = p[ya * s + xb];
  float v10 = p[yb * s + xa], v11 = p[yb * s + xb];
  float v0 = v00 + (v01 - v00) * tx;
  float v1 = v10 + (v11 - v10) * tx;
  return v0 + (v1 - v0) * ty;
}

// ga(y,x) = (grad0 + sum_{l=1..9} upsampled level_l) / 10, evaluated lazily
__device__ float eval_ga(const float* tgt, const float* mout, const float* pyr,
                         int ch, int iy, int ix) {
  int base = ch * HW + iy * WW + ix;
  float sum = -tgt[base] + 0.01f * mout[base];  // level 0 = grad0
#pragma unroll
  for (int l = 1; l < 10; ++l) {
    int s = LSZ[l];
    sum += bilerp(pyr + LOFF[l] + ch * s * s, s, iy, ix);
  }
  return sum * 0.1f;
}

// ---------------- kernels ----------------

__global__ void k_zero(float* p, unsigned long long n4) {  // n4 = #float4
  float4 z = {0.f, 0.f, 0.f, 0.f};
  float4* o = (float4*)p;
  for (unsigned long long i = (unsigned long long)blockIdx.x * blockDim.x + threadIdx.x;
       i < n4; i += (unsigned long long)gridDim.x * blockDim.x)
    o[i] = z;
}

__global__ void k_state_init(const float* pts, float* ws) {
  int i = blockIdx.x * 256 + threadIdx.x;  // point index [0,12288)
  ws[WS_P + 2 * i]     = pts[2 * i];
  ws[WS_P + 2 * i + 1] = pts[2 * i + 1];
  ws[WS_SQ + 2 * i]     = 0.f;
  ws[WS_SQ + 2 * i + 1] = 0.f;
  if (i == 0) ((int*)ws)[WS_DONE] = 0;
}

__global__ void k_scatter(const float* pts, float* mask, float* pcopy) {
  int i = blockIdx.x * 256 + threadIdx.x;
  float px = pts[2 * i], py = pts[2 * i + 1];
  int ix = rnd_clamp(px), iy = rnd_clamp(py);
  int ch = i / NPTS;
  mask[(size_t)ch * HW + iy * WW + ix] = 1.0f;
  if (pcopy) { pcopy[2 * i] = px; pcopy[2 * i + 1] = py; }
}

// level0 -> level1: TDM tile load of masks_target into LDS, then 3x3/9 stride-2
__global__ void __launch_bounds__(256) k_down1(const float* tgt, float* pyr) {
  __shared__ float lds_t[65 * 129];
  int ch   = blockIdx.x >> 5;
  int tile = blockIdx.x & 31;
  int oy0 = (tile >> 2) * 32;   // 8 row-tiles
  int ox0 = (tile & 3) * 64;    // 4 col-tiles
  int ystart = max(2 * oy0 - 1, 0);
  int xstart = max(2 * ox0 - 1, 0);
  int hclip = (2 * oy0 + 63) - ystart + 1;    // 64 or 65
  int wclip = (2 * ox0 + 127) - xstart + 1;   // 128 or 129
  const float* src = tgt + (size_t)ch * HW + (size_t)ystart * WW + xstart;

#if __has_builtin(__builtin_amdgcn_tensor_load_to_lds)
  if (threadIdx.x < 32) {  // one wave issues the TDM DMA
    uint32_t lds_addr = (uint32_t)(uintptr_t)(void*)&lds_t[0];
    uint64_t ga = (uint64_t)(uintptr_t)src;
    // D# group0: count=1, lds_addr[63:32], global_addr[120:64], type=2
    v4u g0 = {1u, lds_addr, (uint32_t)ga,
              (uint32_t)((ga >> 32) & 0x01FFFFFFu) | (2u << 30)};
    // D# group1: data_size=2 (4B); tensor_dim0=wclip, tensor_dim1=hclip;
    // tile_dim0=wclip, tile_dim1=hclip, tile_dim2=0; stride0=512, stride1=HW
    v8i g1 = {(int)(2u << 16),
              (int)(((uint32_t)wclip & 0xFFFFu) << 16),
              (int)(((uint32_t)hclip & 0xFFFFu) << 16),
              (int)(((uint32_t)wclip & 0xFFFFu) << 16),
              (int)((uint32_t)hclip & 0xFFFFu),
              (int)WW, 0, (int)(HW >> 16)};
    v4i gz4 = {0, 0, 0, 0};
    v8i gz8 = {0, 0, 0, 0, 0, 0, 0, 0};
    // clang-23 / therock signature: (g0, g1, g2, g3, g4, cpol)
    __builtin_amdgcn_tensor_load_to_lds(g0, g1, gz4, gz4, gz8, 0);
    __builtin_amdgcn_s_wait_tensorcnt(0);
  }
#else
  for (int idx = threadIdx.x; idx < hclip * wclip; idx += 256) {
    int yy = idx / wclip, xx = idx - yy * wclip;
    lds_t[idx] = src[(size_t)yy * WW + xx];
  }
#endif
  __syncthreads();

  float* lvl1 = pyr + LOFF[1] + ch * 256 * 256;
#pragma unroll
  for (int k = 0; k < 8; ++k) {
    int idx = threadIdx.x + k * 256;
    int oy = oy0 + (idx >> 6);
    int ox = ox0 + (idx & 63);
    float acc = 0.f;
#pragma unroll
    for (int dy = -1; dy <= 1; ++dy) {
      int y = 2 * oy + dy;
#pragma unroll
      for (int dx = -1; dx <= 1; ++dx) {
        int x = 2 * ox + dx;
        float v = (y < 0 || x < 0) ? OOBV
                                   : -lds_t[(y - ystart) * wclip + (x - xstart)];
        acc += v;
      }
    }
    lvl1[oy * 256 + ox] = acc * (1.0f / 9.0f);
  }
}

// inject 0.01*masks_out correction into level1 (dedup coincident points)
__global__ void __launch_bounds__(128) k_fix1(const float* pts, float* pyr) {
  __shared__ int sy[128], sx[128];
  int ch = blockIdx.x;
  int n = threadIdx.x;
  int i = ch * NPTS + n;
  int ix = rnd_clamp(pts[2 * i]);
  int iy = rnd_clamp(pts[2 * i + 1]);
  sy[n] = iy; sx[n] = ix;
  __syncthreads();
  for (int j = 0; j < n; ++j)
    if (sy[j] == iy && sx[j] == ix) return;  // not first at this pixel
  float* lvl1 = pyr + LOFF[1] + ch * 256 * 256;
  const float add = 0.01f / 9.0f;
  int ky = iy >> 1, kx = ix >> 1;
  int ny = (iy & 1) ? 2 : 1;
  int nx = (ix & 1) ? 2 : 1;
  for (int a = 0; a < ny; ++a) {
    int cy = ky + a; if (cy > 255) continue;
    for (int b = 0; b < nx; ++b) {
      int cx = kx + b; if (cx > 255) continue;
      atomicAdd(&lvl1[cy * 256 + cx], add);
    }
  }
}

// generic level l -> l+1 downsample (small levels)
__global__ void k_downL(float* pyr, int l) {
  int ch = blockIdx.x;
  int sin = LSZ[l], sout = LSZ[l + 1];
  const float* in = pyr + LOFF[l] + ch * sin * sin;
  float* out = pyr + LOFF[l + 1] + ch * sout * sout;
  for (int idx = threadIdx.x; idx < sout * sout; idx += blockDim.x) {
    int oy = idx / sout, ox = idx - oy * sout;
    float acc = 0.f;
    for (int dy = -1; dy <= 1; ++dy) {
      int y = 2 * oy + dy;
      for (int dx = -1; dx <= 1; ++dx) {
        int x = 2 * ox + dx;
        acc += (y < 0 || x < 0) ? OOBV : in[y * sin + x];
      }
    }
    out[idx] = acc * (1.f / 9.f);
  }
}

// per-step: block maxima of pl
__global__ void __launch_bounds__(256) k_max(const float* tgt, float* ws) {
  __shared__ float red[256];
  int i = blockIdx.x * 256 + threadIdx.x;
  int ix = rnd_clamp(ws[WS_P + 2 * i]);
  int iy = rnd_clamp(ws[WS_P + 2 * i + 1]);
  int ch = i / NPTS, n = i % NPTS;
  float t = tgt[(size_t)ch * HW + iy * WW + ix];
  float pl = (1.0f - t) * __expf(-(float)n * (10.0f / 128.0f));
  red[threadIdx.x] = pl;
  __syncthreads();
  for (int s = 128; s > 0; s >>= 1) {
    if (threadIdx.x < s)
      red[threadIdx.x] = fmaxf(red[threadIdx.x], red[threadIdx.x + s]);
    __syncthreads();
  }
  if (threadIdx.x == 0) ws[WS_BMAX + blockIdx.x] = red[0];
}

// per-step: gated RMSprop update with lazy gradient-surface gathers
__global__ void __launch_bounds__(256) k_update(const float* tgt, const float* mout,
                                                const float* pyr, float* ws) {
  __shared__ float s_max;
  __shared__ int s_done;
  if (threadIdx.x == 0) {
    float m = ws[WS_BMAX];
    for (int j = 1; j < 48; ++j) m = fmaxf(m, ws[WS_BMAX + j]);
    s_max = m;
    s_done = ((int*)ws)[WS_DONE];
  }
  __syncthreads();
  bool cont = s_max > 0.1f;
  if (blockIdx.x == 0 && threadIdx.x == 0 && !cont) ((int*)ws)[WS_DONE] = 1;
  if (s_done != 0 || !cont) return;

  int i = blockIdx.x * 256 + threadIdx.x;
  float px = ws[WS_P + 2 * i], py = ws[WS_P + 2 * i + 1];
  float sq0 = ws[WS_SQ + 2 * i], sq1 = ws[WS_SQ + 2 * i + 1];
  int ix = rnd_clamp(px), iy = rnd_clamp(py);
  int ch = i / NPTS, n = i % NPTS;
  float t = tgt[(size_t)ch * HW + iy * WW + ix];
  float pl = (1.0f - t) * __expf(-(float)n * (10.0f / 128.0f));

  float ga_u = eval_ga(tgt, mout, pyr, ch, max(iy - 1, 0), ix);
  float ga_d = eval_ga(tgt, mout, pyr, ch, min(iy + 1, HH - 1), ix);
  float ga_l = eval_ga(tgt, mout, pyr, ch, iy, max(ix - 1, 0));
  float ga_r = eval_ga(tgt, mout, pyr, ch, iy, min(ix + 1, WW - 1));
  float gxp = (ga_u - ga_d) * 0.5f;   // J[0]: central diff along H
  float gyp = (ga_l - ga_r) * 0.5f;   // J[1]: central diff along W

  float g0 = -gyp * pl;   // applied to x (p[...,0])
  float g1 = -gxp * pl;   // applied to y (p[...,1])
  sq0 = ALPHAV * sq0 + (1.f - ALPHAV) * g0 * g0;
  sq1 = ALPHAV * sq1 + (1.f - ALPHAV) * g1 * g1;
  px -= LRATE * g0 / (sqrtf(sq0) + EPSV);
  py -= LRATE * g1 / (sqrtf(sq1) + EPSV);
  ws[WS_P + 2 * i] = px;  ws[WS_P + 2 * i + 1] = py;
  ws[WS_SQ + 2 * i] = sq0; ws[WS_SQ + 2 * i + 1] = sq1;
}

extern "C" void kernel_launch(void* const* d_in, const int* in_sizes, int n_in,
                              void* d_out, int out_size, void* d_ws, size_t ws_size,
                              hipStream_t stream) {
  const float* pts = (const float*)d_in[0];   // (B,V,N,2)
  const float* tgt = (const float*)d_in[1];   // (B,V,H,W)
  float* out = (float*)d_out;
  float* mask_out = out;
  float* mask_opt = out + MASK_ELEMS;         // doubles as pyramid scratch
  float* p_opt    = out + 2 * MASK_ELEMS;
  float* ws  = (float*)d_ws;
  float* pyr = mask_opt;

  // 1) zero masks_out; init point state
  k_zero<<<4096, 256, 0, stream>>>(mask_out, (unsigned long long)(MASK_ELEMS / 4));
  k_state_init<<<48, 256, 0, stream>>>(pts, ws);
  // 2) scatter masks_out (original points)
  k_scatter<<<48, 256, 0, stream>>>(pts, mask_out, nullptr);
  // 3) pyramid: level1 via TDM tiles, sparse fix, then chain
  k_down1<<<CHN * 32, 256, 0, stream>>>(tgt, pyr);
  k_fix1<<<CHN, 128, 0, stream>>>(pts, pyr);
  for (int l = 1; l < 9; ++l) k_downL<<<CHN, 256, 0, stream>>>(pyr, l);
  // 4) 10 RMSprop steps with global-max gating
  for (int s = 0; s < 10; ++s) {
    k_max<<<48, 256, 0, stream>>>(tgt, ws);
    k_update<<<48, 256, 0, stream>>>(tgt, mask_out, pyr, ws);
  }
  // 5) finalize: wipe pyramid scratch -> masks_opt, scatter, emit p_opt
  k_zero<<<4096, 256, 0, stream>>>(mask_opt, (unsigned long long)(MASK_ELEMS / 4));
  k_scatter<<<48, 256, 0, stream>>>(ws + WS_P, mask_opt, p_opt);
}